// SudokuSolver_47991964565669
// MI455X (gfx1250) — compile-verified
//
#include <hip/hip_runtime.h>
#include <stdint.h>

typedef float v2f __attribute__((ext_vector_type(2)));
typedef float v8f __attribute__((ext_vector_type(8)));

#define CELLS 81
#define NDIG  9
#define HID   100
#define K1P   28    // 27 padded to 7 K-slabs of 4
#define N1P   112   // 100 padded to 7 N-tiles of 16
#define M1P   32    // 27 padded to 2 M-tiles of 16

static __device__ __forceinline__ unsigned long long umax64(unsigned long long a,
                                                            unsigned long long b) {
  return a > b ? a : b;
}

__global__ __launch_bounds__(256)
void sudoku_solver_gfx1250(const float* __restrict__ x,
                           const float* __restrict__ W1,
                           const float* __restrict__ W2,
                           float* __restrict__ out_pred,
                           float* __restrict__ out_final)
{
  __shared__ float xc[CELLS * NDIG];      // current board state
  __shared__ float xp[CELLS * NDIG];      // prediction output
  __shared__ float B1[K1P * N1P];         // W1^T padded: B1[k][h]
  __shared__ float B2[HID * 16];          // W2^T padded: B2[k][d]
  __shared__ float Amat[M1P * K1P];       // block-diagonal group sums (27x27 logical)
  __shared__ float G[M1P * N1P];          // G = Amat @ B1 (row/col/box partial hiddens)
  __shared__ float Y[96 * 16];            // logits, padded
  __shared__ unsigned long long key[128]; // argmax reduction keys
  __shared__ int digitArr[CELLS];
  __shared__ int emptyF[CELLS];
  __shared__ int Esh;

  const int tid  = threadIdx.x;
  const int wave = tid >> 5;
  const int lane = tid & 31;
  const int n16  = lane & 15;   // M/N index within a 16-wide tile
  const int kh   = lane >> 4;   // K-half selector for f32 WMMA fragments
  const int b    = blockIdx.x;

  const float* xb = x + (size_t)b * (CELLS * NDIG);

  // ---- stage board + weights into LDS ----
  for (int i = tid; i < CELLS * NDIG; i += 256) { float v = xb[i]; xc[i] = v; xp[i] = v; }
  for (int i = tid; i < K1P * N1P; i += 256) {
    int k = i / N1P, h = i - k * N1P;
    B1[i] = (k < 27 && h < HID) ? W1[h * 27 + k] : 0.0f;
  }
  for (int i = tid; i < HID * 16; i += 256) {
    int k = i >> 4, d = i & 15;
    B2[i] = (d < NDIG) ? W2[d * HID + k] : 0.0f;
  }
  for (int i = tid; i < M1P * K1P; i += 256) Amat[i] = 0.0f;  // zeros persist off-diagonal
  __syncthreads();

  // ---- initial empty flags + per-board step count ----
  if (tid < CELLS) {
    float s = 0.f;
    for (int d = 0; d < NDIG; ++d) s += xc[tid * NDIG + d];
    emptyF[tid] = (s == 0.0f) ? 1 : 0;
  }
  __syncthreads();
  if (tid == 0) {
    int e = 0;
    for (int q = 0; q < CELLS; ++q) e += emptyF[q];
    Esh = e;
  }
  __syncthreads();
  const int E = Esh;  // extra steps beyond E are no-ops in the reference

  for (int step = 0; step < E; ++step) {
    // ---- (a) constraint aggregation: 27 group-sum vectors, block-diagonal A ----
    for (int idx = tid; idx < 243; idx += 256) {
      int g = idx / 81, rem = idx - g * 81;
      int i9 = rem / 9, d = rem - i9 * 9;
      float s = 0.f;
      if (g == 0) {                 // row sums
        for (int c2 = 0; c2 < 9; ++c2) s += xc[(i9 * 9 + c2) * 9 + d];
        Amat[i9 * K1P + d] = s;
      } else if (g == 1) {          // column sums
        for (int r2 = 0; r2 < 9; ++r2) s += xc[(r2 * 9 + i9) * 9 + d];
        Amat[(9 + i9) * K1P + (9 + d)] = s;
      } else {                      // box sums
        int br = (i9 / 3) * 3, bc = (i9 % 3) * 3;
        for (int rr = 0; rr < 3; ++rr)
          for (int cc = 0; cc < 3; ++cc)
            s += xc[((br + rr) * 9 + (bc + cc)) * 9 + d];
        Amat[(18 + i9) * K1P + (18 + d)] = s;
      }
    }
    __syncthreads();

    // ---- (b1) GEMM1: G(32x112) = Amat(32x28) @ B1(28x112), 14 tiles over 8 waves ----
    for (int t = wave; t < 14; t += 8) {
      int mt = t / 7, nt = t - mt * 7;
      v8f acc = {0.f, 0.f, 0.f, 0.f, 0.f, 0.f, 0.f, 0.f};
      for (int ks = 0; ks < 7; ++ks) {
        int k0 = ks * 4 + kh * 2;
        v2f a, bb;
        a.x = Amat[(mt * 16 + n16) * K1P + k0];
        a.y = Amat[(mt * 16 + n16) * K1P + k0 + 1];
        bb.x = B1[k0 * N1P + nt * 16 + n16];
        bb.y = B1[(k0 + 1) * N1P + nt * 16 + n16];
        acc = __builtin_amdgcn_wmma_f32_16x16x4_f32(false, a, false, bb,
                                                    (short)0, acc, false, false);
      }
      for (int r = 0; r < 8; ++r)
        G[(mt * 16 + kh * 8 + r) * N1P + nt * 16 + n16] = acc[r];
    }
    __syncthreads();

    // ---- (b2) GEMM2: Y(96x16) = relu(G[row]+G[col]+G[box]) @ B2(100x16) ----
    // A operand assembled on the fly; h never materialized.
    if (wave < 6) {
      int mt = wave;
      int q = mt * 16 + n16;
      bool valid = q < CELLS;
      int r0 = 0, c0 = 0, b0 = 0;
      if (valid) {
        int rr = q / 9, cc = q - rr * 9;
        r0 = rr; c0 = 9 + cc; b0 = 18 + (rr / 3) * 3 + cc / 3;
      }
      v8f acc = {0.f, 0.f, 0.f, 0.f, 0.f, 0.f, 0.f, 0.f};
      for (int ks = 0; ks < 25; ++ks) {   // K = 100 exactly
        int j = ks * 4 + kh * 2;
        float a0 = 0.f, a1 = 0.f;
        if (valid) {
          a0 = G[r0 * N1P + j]     + G[c0 * N1P + j]     + G[b0 * N1P + j];
          a1 = G[r0 * N1P + j + 1] + G[c0 * N1P + j + 1] + G[b0 * N1P + j + 1];
          a0 = fmaxf(a0, 0.f);
          a1 = fmaxf(a1, 0.f);
        }
        v2f a, bb;
        a.x = a0; a.y = a1;
        bb.x = B2[j * 16 + n16];
        bb.y = B2[(j + 1) * 16 + n16];
        acc = __builtin_amdgcn_wmma_f32_16x16x4_f32(false, a, false, bb,
                                                    (short)0, acc, false, false);
      }
      for (int r = 0; r < 8; ++r)
        Y[(mt * 16 + kh * 8 + r) * 16 + n16] = acc[r];
    }
    __syncthreads();

    // ---- (c) per-cell score (max of softmax) + first-occurrence argmax key ----
    if (tid < 128) key[tid] = 0ull;
    if (tid < CELLS && emptyF[tid]) {
      int q = tid;
      float m = Y[q * 16 + 0];
      int dig = 0;
      for (int d = 1; d < NDIG; ++d) {
        float v = Y[q * 16 + d];
        if (v > m) { m = v; dig = d; }   // strict '>' == first max, like jnp.argmax
      }
      float Z = 0.f;
      for (int d = 0; d < NDIG; ++d) Z += __expf(Y[q * 16 + d] - m);
      float score = 1.0f / Z;            // == max_d softmax(y)_d
      digitArr[q] = dig;
      key[q] = ((unsigned long long)__float_as_uint(score) << 32)
             | (unsigned long long)(0xFFFFFFFFu - (unsigned)q);  // tie -> lowest q
    }
    __syncthreads();

    // ---- wave0 max-reduction over 128 keys, then commit the chosen cell ----
    if (tid < 32) {
      unsigned long long kk = key[lane];
      kk = umax64(kk, key[lane + 32]);
      kk = umax64(kk, key[lane + 64]);
      kk = umax64(kk, key[lane + 96]);
      for (int off = 16; off > 0; off >>= 1)
        kk = umax64(kk, __shfl_xor(kk, off, 32));
      if (lane == 0) {
        int q = (int)(0xFFFFFFFFu - (unsigned)(kk & 0xFFFFFFFFull));
        int dig = digitArr[q];
        xc[q * NDIG + dig] = 1.0f;
        emptyF[q] = 0;
        // xp row for the chosen cell = softmax(Y[q]) (its final reference value)
        float m = Y[q * 16 + 0];
        for (int d = 1; d < NDIG; ++d) m = fmaxf(m, Y[q * 16 + d]);
        float e[NDIG]; float Z = 0.f;
        for (int d = 0; d < NDIG; ++d) { e[d] = __expf(Y[q * 16 + d] - m); Z += e[d]; }
        float inv = 1.0f / Z;
        for (int d = 0; d < NDIG; ++d) xp[q * NDIG + d] = e[d] * inv;
      }
    }
    __syncthreads();
  }

  // ---- write outputs: (x_pred, x_final) concatenated ----
  float* op = out_pred  + (size_t)b * (CELLS * NDIG);
  float* of = out_final + (size_t)b * (CELLS * NDIG);
  for (int i = tid; i < CELLS * NDIG; i += 256) {
    op[i] = xp[i];
    of[i] = xc[i];
  }
}

extern "C" void kernel_launch(void* const* d_in, const int* in_sizes, int n_in,
                              void* d_out, int out_size, void* d_ws, size_t ws_size,
                              hipStream_t stream) {
  (void)n_in; (void)d_ws; (void)ws_size; (void)out_size;
  const float* x  = (const float*)d_in[0];
  // d_in[1] is the constraint mask c — recomputed analytically on device.
  const float* W1 = (const float*)d_in[2];
  const float* W2 = (const float*)d_in[3];
  const int B = in_sizes[0] / (CELLS * NDIG);   // 1024
  float* out_pred  = (float*)d_out;
  float* out_final = out_pred + (size_t)B * (CELLS * NDIG);
  sudoku_solver_gfx1250<<<B, 256, 0, stream>>>(x, W1, W2, out_pred, out_final);
}